// DIFF_GraphAttention_58969900974822
// MI455X (gfx1250) — compile-verified
//
#include <hip/hip_runtime.h>
#include <math.h>
#include <stdint.h>

typedef float v2f __attribute__((ext_vector_type(2)));
typedef float v4f __attribute__((ext_vector_type(4)));
typedef float v8f __attribute__((ext_vector_type(8)));

#define FDIM 128
#define DEG 32
#define WPB 4            // waves per block (1 node per wave)
#define RSTR 132         // LDS row stride in dwords (padded vs 128 to break bank conflicts)

// ---------------- kernel 1: x = tanh(features), vectorized ----------------
__global__ void tanh_kernel(const float* __restrict__ in, float* __restrict__ out, int n4) {
  int i = blockIdx.x * blockDim.x + threadIdx.x;
  int stride = gridDim.x * blockDim.x;
  for (; i < n4; i += stride) {
    v4f v = ((const v4f*)in)[i];
    v4f r;
    r.x = tanhf(v.x); r.y = tanhf(v.y); r.z = tanhf(v.z); r.w = tanhf(v.w);
    ((v4f*)out)[i] = r;
  }
}

__device__ __forceinline__ float wave_sum(float v) {
  #pragma unroll
  for (int m = 16; m >= 1; m >>= 1) v += __shfl_xor(v, m, 32);
  return v;
}
__device__ __forceinline__ float wave_max(float v) {
  #pragma unroll
  for (int m = 16; m >= 1; m >>= 1) v = fmaxf(v, __shfl_xor(v, m, 32));
  return v;
}

// ---------------- kernel 2: per-node graph attention ----------------
__global__ void __launch_bounds__(WPB * 32)
gat_kernel(const float* __restrict__ xt,      // [N,128] = tanh(features)
           const int*   __restrict__ cols,    // [N*32] sorted neighbor ids per node
           const float* __restrict__ high,    // [128]
           const float* __restrict__ diff,    // [128]
           float* __restrict__ out,           // [N,128]
           int N)
{
  __shared__ float ldsNh[WPB * DEG * RSTR];   // staged neighbor tiles
  __shared__ float wlds[FDIM];                // combined attention vector

  const int lane = threadIdx.x & 31;
  const int wave = threadIdx.x >> 5;
  int node = blockIdx.x * WPB + wave;
  const bool active = node < N;
  if (!active) node = N - 1;                  // clamp: keep memory ops valid, skip store

  float* Nh = ldsNh + wave * (DEG * RSTR);

  // stage w = high - 0.5*diff into LDS (all waves write identical data; benign)
  {
    v4f hv = *(const v4f*)(high + lane * 4);
    v4f dv = *(const v4f*)(diff + lane * 4);
    v4f wv;
    wv.x = hv.x - 0.5f * dv.x;
    wv.y = hv.y - 0.5f * dv.y;
    wv.z = hv.z - 0.5f * dv.z;
    wv.w = hv.w - 0.5f * dv.w;
    *(v4f*)(wlds + lane * 4) = wv;
  }

  // ---- gather via CDNA5 async copy: global -> LDS with no VGPR round-trip ----
  // Row j (512B contiguous) is fetched coalesced by the whole wave directly into
  // LDS; 32 ops in flight tracked by ASYNCcnt.
  int colv = cols[(size_t)node * DEG + lane];
  uint32_t ldsbase = (uint32_t)(uintptr_t)(Nh);   // flat LDS addr truncates to LDS offset
  uint32_t voff = (uint32_t)lane * 16u;
  #pragma unroll
  for (int j = 0; j < DEG; ++j) {
    int c = __builtin_amdgcn_readlane(colv, j);   // scalar neighbor id
    uint64_t gaddr = (uint64_t)(uintptr_t)(xt + (size_t)c * FDIM);  // SGPR-pair row base
    uint32_t ldsaddr = ldsbase + (uint32_t)j * (RSTR * 4u) + voff;
    asm volatile("global_load_async_to_lds_b128 %0, %1, %2"
                 :: "v"(ldsaddr), "v"(voff), "s"(gaddr)
                 : "memory");
  }
  asm volatile("s_wait_asynccnt 0x0" ::: "memory");
  __syncthreads();

  // scores: lane j = dot(Nh[j], w)
  float s = 0.f;
  #pragma unroll
  for (int d = 0; d < FDIM / 4; ++d) {
    v4f r  = *(const v4f*)(Nh + lane * RSTR + d * 4);
    v4f wv = *(const v4f*)(wlds + d * 4);          // broadcast (same addr all lanes)
    s += r.x * wv.x + r.y * wv.y + r.z * wv.z + r.w * wv.w;
  }

  // stable softmax over the 32 edge scores (one per lane)
  float mmax = wave_max(s);
  float e    = expf(s - mmax);
  float ssum = wave_sum(e);
  float att  = e / ssum;

  // weighted aggregation out[0:128] = att[0:32] . Nh[0:32][0:128]
  // via chained V_WMMA_F32_16X16X4_F32 (full fp32): 8 K-chunks x 8 column tiles.
  // A[m][k] = att[4*kc+k] for all m (broadcast down M); B[k][n] = Nh[4*kc+k][16*t+n].
  const int kh  = (lane >= 16) ? 2 : 0;   // K-half split per ISA A/B VGPR layout
  const int col = lane & 15;

  v8f acc[8] = {};
  #pragma unroll
  for (int kc = 0; kc < 8; ++kc) {
    v2f a;
    a.x = __shfl(att, 4 * kc + kh,     32);
    a.y = __shfl(att, 4 * kc + kh + 1, 32);
    const float* r0 = Nh + (4 * kc + kh) * RSTR + col;
    #pragma unroll
    for (int t = 0; t < 8; ++t) {
      v2f b;
      b.x = r0[t * 16];            // row 4kc+kh
      b.y = r0[RSTR + t * 16];     // row 4kc+kh+1
      acc[t] = __builtin_amdgcn_wmma_f32_16x16x4_f32(
          /*neg_a=*/false, a, /*neg_b=*/false, b,
          /*c_mod=*/(short)0, acc[t], /*reuse_a=*/false, /*reuse_b=*/false);
    }
  }

  // Every D row is identical (A broadcast down M). Lanes 0..15 hold M=0 (N=lane)
  // in VGPR0; lanes 16..31 hold M=8 (N=lane-16), same values. Pack the wave:
  // lower half stores tiles 0..3, upper half stores tiles 4..7 -> 4 full-wave stores.
  if (active) {
    float* op = out + (size_t)node * FDIM;
    const int thi = (lane >= 16) ? 4 : 0;
    #pragma unroll
    for (int t = 0; t < 4; ++t) {
      float v = (lane >= 16) ? acc[t + 4][0] : acc[t][0];
      op[(t + thi) * 16 + col] = tanhf(v);
    }
  }
}

// ---------------- host launcher ----------------
extern "C" void kernel_launch(void* const* d_in, const int* in_sizes, int n_in,
                              void* d_out, int out_size, void* d_ws, size_t ws_size,
                              hipStream_t stream) {
  const float* features = (const float*)d_in[0];   // [N,128]
  const int*   adj      = (const int*)  d_in[1];   // [2,E]
  const float* high     = (const float*)d_in[2];   // [1,128]
  const float* diff     = (const float*)d_in[3];   // [1,128]
  float* out = (float*)d_out;

  const int ND = in_sizes[0];
  const int N  = ND / FDIM;
  const int E  = in_sizes[1] / 2;
  const int* cols = adj + E;                        // second row of adj_nei

  float* xt = (float*)d_ws;                         // N*128 floats of scratch

  const int n4 = ND / 4;
  int tblocks = (n4 + 255) / 256;
  if (tblocks > 16384) tblocks = 16384;
  tanh_kernel<<<tblocks, 256, 0, stream>>>(features, xt, n4);

  const int gblocks = (N + WPB - 1) / WPB;
  gat_kernel<<<gblocks, WPB * 32, 0, stream>>>(xt, cols, high, diff, out, N);
}